// SleepStageModel_3461743640608
// MI455X (gfx1250) — compile-verified
//
#include <hip/hip_runtime.h>

// CRF mean NLL on gfx1250 (MI455X).
//
// Denominator (log-partition) in scaled-probability domain:
//   state alpha[j] kept normalized (max_j alpha = 1), per-row log2 scale c2.
//   step: q = Et * alpha (5x5 matvec via 2x V_WMMA_F32_16X16X4_F32),
//         alpha' = q .* exp2(em * 1/ln2), renorm by row max (1 shfl_xor),
//         c2 += log2(max).
// Numerator fused in the same pass (tags/mask stream + emission gather +
// register-table transition lookups via ds_bpermute/__shfl).
//
// Wave tiling: 1 wave = 16 batch rows.
//   A operand (constant Et = exp(T)^T, padded): 16x4 f32 A-layout
//     lanes 0-15 hold row j=lane, {K=0,K=1}; lanes 16-31 hold {K=2,K=3}.
//   B operand (state): 4x16 f32 B-layout assumed to mirror the documented
//     16-bit B layout (here K0,K1 in lanes 0-15, K2,K3 in lanes 16-31):
//     vgpr0 = {K0 | K2}, vgpr1 = {K1 | K3}, batch col = lane&15.
//   C/D: vgpr v, lane l -> M = v + 8*(l>=16), N = l&15  (per ISA 7.12.2).
//   Feedback C->B needs only shfl_xor(.,16) for rows 2,3 (A is constant).

#define NTAG 5
#define BATCH 16384
#define SEQL  512
#define NWAVE (BATCH / 16)   // 1024 waves total

typedef float v2f __attribute__((ext_vector_type(2)));
typedef float v8f __attribute__((ext_vector_type(8)));

__device__ __forceinline__ float ex2(float x) { return __builtin_amdgcn_exp2f(x); }
__device__ __forceinline__ float lg2(float x) { return __builtin_amdgcn_logf(x); }
__device__ __forceinline__ float rcpf(float x) { return __builtin_amdgcn_rcpf(x); }

__global__ __launch_bounds__(256) void crf_fwd_kernel(
    const float* __restrict__ em,      // [B, L, 5]
    const float* __restrict__ T,       // [5, 5]
    const float* __restrict__ startT,  // [5]
    const float* __restrict__ endT,    // [5]
    const int* __restrict__ tags,      // [B, L]
    const unsigned char* __restrict__ mask, // [B, L] (bool)
    float* __restrict__ ws)            // [NWAVE * 2] partial (den, num)
{
  constexpr float INVLN2 = 1.4426950408889634f;
  constexpr float LN2    = 0.6931471805599453f;

  const int lane = threadIdx.x & 31;
  const int wave = blockIdx.x * (blockDim.x >> 5) + (threadIdx.x >> 5);
  const bool hi  = lane >= 16;
  const int  jj  = lane & 15;                 // A row (Et row) / batch column
  const long long b = (long long)wave * 16 + jj;

  // ---- constant A operand: Et[j][i] = exp(T[i][j]), rows j>=5 zero ----
  v2f a_lo = {0.f, 0.f}, a_hi = {0.f, 0.f};
  if (jj < NTAG) {
    const int i0 = hi ? 2 : 0, i1 = hi ? 3 : 1;
    a_lo.x = ex2(T[i0 * NTAG + jj] * INVLN2);
    a_lo.y = ex2(T[i1 * NTAG + jj] * INVLN2);
    if (!hi) a_hi.x = ex2(T[4 * NTAG + jj] * INVLN2);   // K=4; K=5..7 stay 0
  }

  // per-lane state-slot tag indices: lo lanes hold rows {0,1,4}, hi {2,3,-}
  const int ja = hi ? 2 : 0;
  const int jb = hi ? 3 : 1;
  const int jc = hi ? 2 : 4;                  // hi value is a dummy (masked off)

  const float ee_a = ex2(endT[ja] * INVLN2);
  const float ee_b = ex2(endT[jb] * INVLN2);
  const float ee_c = hi ? 0.f : ex2(endT[4] * INVLN2);
  const float st_a = startT[ja];
  const float st_b = startT[jb];
  const float st_c = hi ? 0.f : startT[4];

  // register-resident lookup tables for numerator (via __shfl):
  // lanes 0..24: T flat; lanes 25..29: endT
  float tbl = 0.f;
  if (lane < 25)      tbl = T[lane];
  else if (lane < 30) tbl = endT[lane - 25];
  const float stbl = (lane < NTAG) ? startT[lane] : 0.f;

  const float* emb = em   + (size_t)b * SEQL * NTAG;
  const int*   tgb = tags + (size_t)b * SEQL;
  const unsigned char* mkb = mask + (size_t)b * SEQL;

  // ---- t = 0 init ----
  float s_a = ex2((emb[ja] + st_a) * INVLN2);
  float s_b = ex2((emb[jb] + st_b) * INVLN2);
  float s_c = hi ? 0.f : ex2((emb[jc] + st_c) * INVLN2);
  float mloc = fmaxf(fmaxf(s_a, s_b), s_c);
  float m = fmaxf(mloc, __shfl_xor(mloc, 16));
  float r = rcpf(m);
  s_a *= r; s_b *= r; s_c *= r;
  float c2 = lg2(m);

  int   tp = tgb[0];
  float accn = __shfl(stbl, tp) + emb[tp];
  int   last_tag = tp;

  // ---- sequential forward over t = 1..L-1 ----
  for (int t = 1; t < SEQL; ++t) {
    const float* e = emb + (size_t)t * NTAG;
    const float va = e[ja], vb = e[jb], vc = e[jc];
    const int   tc = tgb[t];
    const int   mt = (int)mkb[t];
    const float eg = e[tc];

    // q = Et * alpha : two chained f32 WMMAs (K = 4..7 part, then K = 0..3)
    v2f b_lo = {s_a, s_b};
    v2f b_hi = {s_c, 0.f};
    v8f q = {};
    q = __builtin_amdgcn_wmma_f32_16x16x4_f32(false, a_hi, false, b_hi,
                                              (short)0, q, false, false);
    q = __builtin_amdgcn_wmma_f32_16x16x4_f32(false, a_lo, false, b_lo,
                                              (short)0, q, false, false);

    // C -> B feedback: rows 2,3 cross to the upper lane half
    const float x2 = __shfl_xor(q[2], 16);
    const float x3 = __shfl_xor(q[3], 16);
    const float n_a = hi ? x2 : q[0];
    const float n_b = hi ? x3 : q[1];

    // multiply by exp(emissions), renormalize
    const float g_a = ex2(va * INVLN2);
    const float g_b = ex2(vb * INVLN2);
    const float g_c = ex2(vc * INVLN2);
    s_a = n_a * g_a;
    s_b = n_b * g_b;
    s_c = hi ? 0.f : q[4] * g_c;
    mloc = fmaxf(fmaxf(s_a, s_b), s_c);
    m = fmaxf(mloc, __shfl_xor(mloc, 16));
    r = rcpf(m);
    s_a *= r; s_b *= r; s_c *= r;
    c2 += lg2(m);

    // fused numerator
    const float tr = __shfl(tbl, tp * NTAG + tc);
    accn = fmaf((float)mt, tr + eg, accn);
    last_tag = mt ? tc : last_tag;
    tp = tc;
  }

  // ---- epilogue: end transitions, per-row log-den, wave reduction ----
  float d = s_a * ee_a + s_b * ee_b + s_c * ee_c;
  d += __shfl_xor(d, 16);
  const float logden = (c2 + lg2(d)) * LN2;     // valid on lanes 0..15
  accn += __shfl(tbl, 25 + last_tag);

  float vden = hi ? 0.f : logden;
  float vnum = hi ? 0.f : accn;
  #pragma unroll
  for (int o = 16; o > 0; o >>= 1) {
    vden += __shfl_xor(vden, o);
    vnum += __shfl_xor(vnum, o);
  }
  if (lane == 0) {
    ws[2 * wave + 0] = vden;
    ws[2 * wave + 1] = vnum;
  }
}

__global__ __launch_bounds__(256) void crf_reduce_kernel(
    const float* __restrict__ ws, float* __restrict__ out)
{
  __shared__ float sden[256];
  __shared__ float snum[256];
  const int t = threadIdx.x;
  float d = 0.f, n = 0.f;
  for (int k = t; k < NWAVE; k += 256) {      // fixed order -> deterministic
    d += ws[2 * k + 0];
    n += ws[2 * k + 1];
  }
  sden[t] = d; snum[t] = n;
  __syncthreads();
  for (int s = 128; s > 0; s >>= 1) {
    if (t < s) { sden[t] += sden[t + s]; snum[t] += snum[t + s]; }
    __syncthreads();
  }
  if (t == 0) out[0] = (sden[0] - snum[0]) * (1.0f / (float)BATCH);
}

extern "C" void kernel_launch(void* const* d_in, const int* in_sizes, int n_in,
                              void* d_out, int out_size, void* d_ws, size_t ws_size,
                              hipStream_t stream) {
  const float* emissions = (const float*)d_in[0];
  const float* transitions = (const float*)d_in[1];
  const float* start_tr = (const float*)d_in[2];
  const float* end_tr = (const float*)d_in[3];
  const int* tags = (const int*)d_in[4];
  const unsigned char* mask = (const unsigned char*)d_in[5];
  float* out = (float*)d_out;
  float* ws = (float*)d_ws;        // needs NWAVE*2*4 = 8 KB

  // 1024 waves total: 128 blocks x 256 threads (8 waves/block)
  crf_fwd_kernel<<<NWAVE / 8, 256, 0, stream>>>(
      emissions, transitions, start_tr, end_tr, tags, mask, ws);
  crf_reduce_kernel<<<1, 256, 0, stream>>>(ws, out);
}